// GIPA2Conv_34119220199760
// MI455X (gfx1250) — compile-verified
//
#include <hip/hip_runtime.h>
#include <math.h>

// Problem constants (match reference)
#define N_NODES  100000
#define N_EDGES  1600000
#define F_NODEC  128
#define F_EDGEC  8
#define OUT_F    64
#define H_HEADS  4
#define D_HEAD   16
#define EPS_F    1e-9f
#define NEG_SLOPE 0.2f
#define SCAN_T   1024

typedef float v2f __attribute__((ext_vector_type(2)));
typedef float v8f __attribute__((ext_vector_type(8)));

// ---------------------------------------------------------------------------
// Kernel 1: fused node GEMMs.  O = X @ W^T for four [64,128] weight matrices.
// One wave (32 threads) per 16x16 output tile.  WMMA f32 16x16x4, K-loop=32.
// ---------------------------------------------------------------------------
__global__ __launch_bounds__(32) void node_gemm_kernel(
    const float* __restrict__ X,
    const float* __restrict__ W0, const float* __restrict__ W1,
    const float* __restrict__ W2, const float* __restrict__ W3,
    float* __restrict__ O0, float* __restrict__ O1,
    float* __restrict__ O2, float* __restrict__ O3)
{
  const int rb   = blockIdx.x;      // node row-block (16 rows)
  const int t    = blockIdx.y;      // 0..15 : which 16-col tile of the 4x64 outputs
  const int lane = threadIdx.x;

  const float* W;
  float* O;
  switch (t >> 2) {
    case 0:  W = W0; O = O0; break;
    case 1:  W = W1; O = O1; break;
    case 2:  W = W2; O = O2; break;
    default: W = W3; O = O3; break;
  }
  const int n0 = (t & 3) * 16;
  const int r0 = rb * 16;

  const int l15   = lane & 15;
  const int khalf = (lane >> 4) * 2;

  const float* aptr = X + (size_t)(r0 + l15) * F_NODEC + khalf;
  const float* bptr = W + (size_t)(n0 + l15) * F_NODEC + khalf;

  v8f c = {0.f, 0.f, 0.f, 0.f, 0.f, 0.f, 0.f, 0.f};
  #pragma unroll 8
  for (int k0 = 0; k0 < F_NODEC; k0 += 4) {
    v2f a = *(const v2f*)(aptr + k0);
    v2f b = *(const v2f*)(bptr + k0);
    c = __builtin_amdgcn_wmma_f32_16x16x4_f32(
        false, a, false, b, (short)0, c, false, false);
  }

  const int col     = n0 + l15;
  const int rowbase = r0 + ((lane >> 4) << 3);
  #pragma unroll
  for (int r = 0; r < 8; ++r)
    O[(size_t)(rowbase + r) * OUT_F + col] = c[r];
}

// ---------------------------------------------------------------------------
// CSR build step 1: zero the degree counters.
// ---------------------------------------------------------------------------
__global__ __launch_bounds__(256) void zero_deg_kernel(
    int* __restrict__ deg_dst, int* __restrict__ deg_src)
{
  int i = blockIdx.x * 256 + threadIdx.x;
  if (i < N_NODES) { deg_dst[i] = 0; deg_src[i] = 0; }
}

// CSR build step 2: degree histogram (only ~3.2M int atomics total).
__global__ __launch_bounds__(256) void deg_count_kernel(
    const int* __restrict__ src, const int* __restrict__ dst,
    int* __restrict__ deg_dst, int* __restrict__ deg_src)
{
  int e = blockIdx.x * 256 + threadIdx.x;
  if (e < N_EDGES) {
    atomicAdd(&deg_dst[dst[e]], 1);
    atomicAdd(&deg_src[src[e]], 1);
  }
}

// CSR build step 3: exclusive scan of both degree arrays.
// One 1024-thread block per array; each thread scans a contiguous chunk,
// Hillis-Steele combines the 1024 partials in LDS.  Also seeds the cursors.
__global__ __launch_bounds__(SCAN_T) void scan_kernel(
    const int* __restrict__ deg_dst, const int* __restrict__ deg_src,
    int* __restrict__ off_dst, int* __restrict__ off_src,
    int* __restrict__ cur_dst, int* __restrict__ cur_src)
{
  const int* deg = (blockIdx.x == 0) ? deg_dst : deg_src;
  int* off = (blockIdx.x == 0) ? off_dst : off_src;
  int* cur = (blockIdx.x == 0) ? cur_dst : cur_src;

  const int t = threadIdx.x;
  const int CH = (N_NODES + SCAN_T - 1) / SCAN_T;  // 98
  int start = t * CH;
  int end = start + CH; if (end > N_NODES) end = N_NODES;

  int sum = 0;
  for (int i = start; i < end; ++i) sum += deg[i];

  __shared__ int part[SCAN_T];
  part[t] = sum;
  __syncthreads();
  for (int sft = 1; sft < SCAN_T; sft <<= 1) {
    int v = (t >= sft) ? part[t - sft] : 0;
    __syncthreads();
    part[t] += v;
    __syncthreads();
  }
  int run = (t == 0) ? 0 : part[t - 1];
  for (int i = start; i < end; ++i) {
    off[i] = run;
    cur[i] = run;
    run += deg[i];
  }
  if (t == SCAN_T - 1) off[N_NODES] = part[SCAN_T - 1];
}

// CSR build step 4: scatter edge ids into the two adjacency lists.
__global__ __launch_bounds__(256) void scatter_kernel(
    const int* __restrict__ src, const int* __restrict__ dst,
    int* __restrict__ cur_dst, int* __restrict__ cur_src,
    int* __restrict__ eid_dst, int* __restrict__ eid_src)
{
  int e = blockIdx.x * 256 + threadIdx.x;
  if (e < N_EDGES) {
    int p = atomicAdd(&cur_dst[dst[e]], 1);
    eid_dst[p] = e;
    int q = atomicAdd(&cur_src[src[e]], 1);
    eid_src[q] = e;
  }
}

// ---------------------------------------------------------------------------
// Segment softmax stats via ONLINE softmax: one grouped traversal produces
// both m (running max) and s (running rescaled exp-sum).  One node per wave,
// 2 channels per lane; the fixed-endpoint attn row and this lane's two rows
// of W_attn_edge are hoisted out of the edge loop.  All gathers are
// 256B-contiguous rows that live in the 192MB L2.  Zero float atomics.
// e = leaky_relu(attn_fixed[n] + attn_gather[other] + feat_edge @ Wae^T)
// ---------------------------------------------------------------------------
__global__ __launch_bounds__(256) void seg_softmax_kernel(
    const int* __restrict__ off, const int* __restrict__ eid,
    const int* __restrict__ other,
    const float* __restrict__ feat_edge, const float* __restrict__ W_attn_edge,
    const float* __restrict__ attn_fixed, const float* __restrict__ attn_gather,
    float* __restrict__ m_out, float* __restrict__ s_out)
{
  const int lane = threadIdx.x & 31;
  const int wave = threadIdx.x >> 5;
  const int n = blockIdx.x * 8 + wave;
  if (n >= N_NODES) return;
  const int c0 = lane * 2;

  float w[16];
  #pragma unroll
  for (int k = 0; k < 8; ++k) {
    w[k]     = W_attn_edge[(size_t)c0 * 8 + k];
    w[8 + k] = W_attn_edge[(size_t)(c0 + 1) * 8 + k];
  }
  const float fx0 = attn_fixed[(size_t)n * OUT_F + c0];
  const float fx1 = attn_fixed[(size_t)n * OUT_F + c0 + 1];

  float m0 = -INFINITY, m1 = -INFINITY, s0 = 0.f, s1 = 0.f;
  const int beg = off[n], end = off[n + 1];
  for (int i = beg; i < end; ++i) {
    __builtin_prefetch(&eid[i + 64], 0, 0);   // global_prefetch_b8
    int e = eid[i];
    int o = other[e];
    const float* fe = feat_edge + (size_t)e * F_EDGEC;
    float ae0 = 0.f, ae1 = 0.f;
    #pragma unroll
    for (int k = 0; k < F_EDGEC; ++k) {
      float f = fe[k];
      ae0 = fmaf(f, w[k], ae0);
      ae1 = fmaf(f, w[8 + k], ae1);
    }
    const float* gr = attn_gather + (size_t)o * OUT_F + c0;
    float e0 = fx0 + gr[0] + ae0;
    float e1 = fx1 + gr[1] + ae1;
    e0 = e0 > 0.f ? e0 : NEG_SLOPE * e0;
    e1 = e1 > 0.f ? e1 : NEG_SLOPE * e1;
    float mn0 = fmaxf(m0, e0);
    s0 = s0 * __expf(m0 - mn0) + __expf(e0 - mn0);
    m0 = mn0;
    float mn1 = fmaxf(m1, e1);
    s1 = s1 * __expf(m1 - mn1) + __expf(e1 - mn1);
    m1 = mn1;
  }
  if (m0 == -INFINITY) m0 = 0.f;   // empty-segment guard (isfinite in ref)
  if (m1 == -INFINITY) m1 = 0.f;
  m_out[(size_t)n * OUT_F + c0]     = m0;
  m_out[(size_t)n * OUT_F + c0 + 1] = m1;
  s_out[(size_t)n * OUT_F + c0]     = s0;
  s_out[(size_t)n * OUT_F + c0 + 1] = s1;
}

// ---------------------------------------------------------------------------
// Final dst-grouped traversal: a = sqrt(clip(p_dst)*clip(p_src)),
// msg_sum[n] = sum_e feat_src_fc[src[e]] * a  -- accumulated in registers,
// zero float atomics.  Dst-side m,s are hoisted; src-side m,s gathered (L2).
// ---------------------------------------------------------------------------
__global__ __launch_bounds__(256) void aggregate_kernel(
    const int* __restrict__ off_dst, const int* __restrict__ eid_dst,
    const int* __restrict__ src,
    const float* __restrict__ feat_edge, const float* __restrict__ W_attn_edge,
    const float* __restrict__ attn_dst, const float* __restrict__ attn_src,
    const float* __restrict__ m_dst, const float* __restrict__ s_dst,
    const float* __restrict__ m_src, const float* __restrict__ s_src,
    const float* __restrict__ feat_src_fc, float* __restrict__ msg_sum)
{
  const int lane = threadIdx.x & 31;
  const int wave = threadIdx.x >> 5;
  const int n = blockIdx.x * 8 + wave;
  if (n >= N_NODES) return;
  const int c0 = lane * 2;

  float w[16];
  #pragma unroll
  for (int k = 0; k < 8; ++k) {
    w[k]     = W_attn_edge[(size_t)c0 * 8 + k];
    w[8 + k] = W_attn_edge[(size_t)(c0 + 1) * 8 + k];
  }
  const float fx0 = attn_dst[(size_t)n * OUT_F + c0];
  const float fx1 = attn_dst[(size_t)n * OUT_F + c0 + 1];
  const float md0 = m_dst[(size_t)n * OUT_F + c0];
  const float md1 = m_dst[(size_t)n * OUT_F + c0 + 1];
  const float sdv0 = s_dst[(size_t)n * OUT_F + c0];
  const float sdv1 = s_dst[(size_t)n * OUT_F + c0 + 1];
  const float sdinv0 = sdv0 > 0.f ? 1.f / sdv0 : 0.f;
  const float sdinv1 = sdv1 > 0.f ? 1.f / sdv1 : 0.f;

  float acc0 = 0.f, acc1 = 0.f;
  const int beg = off_dst[n], end = off_dst[n + 1];
  for (int i = beg; i < end; ++i) {
    __builtin_prefetch(&eid_dst[i + 64], 0, 0);
    int e = eid_dst[i];
    int sidx = src[e];
    const float* fe = feat_edge + (size_t)e * F_EDGEC;
    float ae0 = 0.f, ae1 = 0.f;
    #pragma unroll
    for (int k = 0; k < F_EDGEC; ++k) {
      float f = fe[k];
      ae0 = fmaf(f, w[k], ae0);
      ae1 = fmaf(f, w[8 + k], ae1);
    }
    const float* gr = attn_src + (size_t)sidx * OUT_F + c0;
    float e0 = fx0 + gr[0] + ae0;
    float e1 = fx1 + gr[1] + ae1;
    e0 = e0 > 0.f ? e0 : NEG_SLOPE * e0;
    e1 = e1 > 0.f ? e1 : NEG_SLOPE * e1;

    float ms0 = m_src[(size_t)sidx * OUT_F + c0];
    float ms1 = m_src[(size_t)sidx * OUT_F + c0 + 1];
    float ss0 = s_src[(size_t)sidx * OUT_F + c0];
    float ss1 = s_src[(size_t)sidx * OUT_F + c0 + 1];

    float pd0 = fmaxf(__expf(e0 - md0) * sdinv0, EPS_F);
    float pd1 = fmaxf(__expf(e1 - md1) * sdinv1, EPS_F);
    float ps0 = fmaxf(__expf(e0 - ms0) / ss0, EPS_F);
    float ps1 = fmaxf(__expf(e1 - ms1) / ss1, EPS_F);
    float a0 = sqrtf(pd0 * ps0);
    float a1 = sqrtf(pd1 * ps1);

    acc0 = fmaf(feat_src_fc[(size_t)sidx * OUT_F + c0],     a0, acc0);
    acc1 = fmaf(feat_src_fc[(size_t)sidx * OUT_F + c0 + 1], a1, acc1);
  }
  msg_sum[(size_t)n * OUT_F + c0]     = acc0;
  msg_sum[(size_t)n * OUT_F + c0 + 1] = acc1;
}

// ---------------------------------------------------------------------------
// Kernel: per-head batch-norm of msg_sum (LDS, stride 68 = conflict-free),
// WMMA GEMM with W_agg [64,64], fused epilogue.  4 waves, 16-col tile each.
// ---------------------------------------------------------------------------
__global__ __launch_bounds__(128) void node_final_kernel(
    const float* __restrict__ msg_sum, const float* __restrict__ scale,
    const float* __restrict__ offset,  const float* __restrict__ W_agg,
    const float* __restrict__ b_agg,   const float* __restrict__ dst_part,
    const float* __restrict__ b_dst,   float* __restrict__ out)
{
  __shared__ float hls[16 * 68];
  const int r0  = blockIdx.x * 16;
  const int tid = threadIdx.x;

  for (int i = tid; i < 16 * 64; i += 128) {
    int r = i >> 6, cidx = i & 63;
    hls[r * 68 + cidx] = msg_sum[(size_t)(r0 + r) * OUT_F + cidx];
  }
  __syncthreads();

  if (tid < 64) {
    int r = tid >> 2, h = tid & 3;
    float* p = &hls[r * 68 + h * D_HEAD];
    float sum = 0.f, sq = 0.f;
    #pragma unroll
    for (int j = 0; j < D_HEAD; ++j) { float v = p[j]; sum += v; sq += v * v; }
    float mean = sum * (1.f / D_HEAD);
    float var  = sq * (1.f / D_HEAD) - mean * mean;
    float rinv = rsqrtf(var + EPS_F);
    #pragma unroll
    for (int j = 0; j < D_HEAD; ++j) {
      float v = p[j];
      p[j] = (v - mean) * scale[h * D_HEAD + j] * rinv + offset[h * D_HEAD + j];
    }
  }
  __syncthreads();

  const int wave = tid >> 5;
  const int lane = tid & 31;
  const int n0 = wave * 16;
  const int l15 = lane & 15;
  const int khalf = (lane >> 4) * 2;

  const float* abase = &hls[l15 * 68 + khalf];
  const float* bbase = W_agg + (size_t)(n0 + l15) * OUT_F + khalf;

  v8f c = {0.f, 0.f, 0.f, 0.f, 0.f, 0.f, 0.f, 0.f};
  #pragma unroll
  for (int k0 = 0; k0 < OUT_F; k0 += 4) {
    v2f a = *(const v2f*)(abase + k0);
    v2f b = *(const v2f*)(bbase + k0);
    c = __builtin_amdgcn_wmma_f32_16x16x4_f32(
        false, a, false, b, (short)0, c, false, false);
  }

  const int col = n0 + l15;
  const int rowbase = r0 + ((lane >> 4) << 3);
  const float badd = b_agg[col] + b_dst[col];
  #pragma unroll
  for (int r = 0; r < 8; ++r) {
    int row = rowbase + r;
    out[(size_t)row * OUT_F + col] =
        c[r] + badd + dst_part[(size_t)row * OUT_F + col];
  }
}

// ---------------------------------------------------------------------------
extern "C" void kernel_launch(void* const* d_in, const int* in_sizes, int n_in,
                              void* d_out, int out_size, void* d_ws, size_t ws_size,
                              hipStream_t stream) {
  const float* feat_src    = (const float*)d_in[0];
  const float* feat_edge   = (const float*)d_in[1];
  const float* W_src       = (const float*)d_in[2];
  const float* W_dst       = (const float*)d_in[3];
  const float* b_dst       = (const float*)d_in[4];
  const float* W_attn_src  = (const float*)d_in[5];
  const float* W_attn_dst  = (const float*)d_in[6];
  const float* W_attn_edge = (const float*)d_in[7];
  const float* scale       = (const float*)d_in[8];
  const float* offset      = (const float*)d_in[9];
  const float* W_agg       = (const float*)d_in[10];
  const float* b_agg       = (const float*)d_in[11];
  const int*   src         = (const int*)d_in[12];
  const int*   dst         = (const int*)d_in[13];

  const size_t NC = (size_t)N_NODES * OUT_F;
  float* ws          = (float*)d_ws;
  float* feat_src_fc = ws + 0 * NC;
  float* attn_src    = ws + 1 * NC;
  float* attn_dst    = ws + 2 * NC;
  float* dst_part    = ws + 3 * NC;
  float* m_dst       = ws + 4 * NC;
  float* m_src       = ws + 5 * NC;
  float* s_dst       = ws + 6 * NC;
  float* s_src       = ws + 7 * NC;
  float* msg_sum     = ws + 8 * NC;

  int* iws     = (int*)(ws + 9 * NC);
  int* off_dst = iws;                       // N+1
  int* off_src = off_dst + (N_NODES + 1);   // N+1
  int* cur_dst = off_src + (N_NODES + 1);   // N
  int* cur_src = cur_dst + N_NODES;         // N
  int* deg_dst = cur_src + N_NODES;         // N
  int* deg_src = deg_dst + N_NODES;         // N
  int* eid_dst = deg_src + N_NODES;         // E
  int* eid_src = eid_dst + N_EDGES;         // E

  // 1) Fused node GEMMs (WMMA f32 16x16x4).
  node_gemm_kernel<<<dim3(N_NODES / 16, 16), 32, 0, stream>>>(
      feat_src, W_src, W_attn_src, W_attn_dst, W_dst,
      feat_src_fc, attn_src, attn_dst, dst_part);

  // 2) Build CSR adjacency for both groupings.
  zero_deg_kernel<<<(N_NODES + 255) / 256, 256, 0, stream>>>(deg_dst, deg_src);
  deg_count_kernel<<<(N_EDGES + 255) / 256, 256, 0, stream>>>(
      src, dst, deg_dst, deg_src);
  scan_kernel<<<2, SCAN_T, 0, stream>>>(
      deg_dst, deg_src, off_dst, off_src, cur_dst, cur_src);
  scatter_kernel<<<(N_EDGES + 255) / 256, 256, 0, stream>>>(
      src, dst, cur_dst, cur_src, eid_dst, eid_src);

  // 3) Grouped online-softmax traversals (no float atomics).
  const unsigned nblocks = (N_NODES + 7) / 8;
  seg_softmax_kernel<<<nblocks, 256, 0, stream>>>(
      off_dst, eid_dst, src, feat_edge, W_attn_edge,
      attn_dst, attn_src, m_dst, s_dst);
  seg_softmax_kernel<<<nblocks, 256, 0, stream>>>(
      off_src, eid_src, dst, feat_edge, W_attn_edge,
      attn_src, attn_dst, m_src, s_src);

  // 4) Register-accumulated message aggregation (no float atomics).
  aggregate_kernel<<<nblocks, 256, 0, stream>>>(
      off_dst, eid_dst, src, feat_edge, W_attn_edge,
      attn_dst, attn_src, m_dst, s_dst, m_src, s_src,
      feat_src_fc, msg_sum);

  // 5) Per-head norm + W_agg GEMM (WMMA) + fused epilogue.
  node_final_kernel<<<N_NODES / 16, 128, 0, stream>>>(
      msg_sum, scale, offset, W_agg, b_agg, dst_part, b_dst, (float*)d_out);
}